// Glove_61787399521004
// MI455X (gfx1250) — compile-verified
//
#include <hip/hip_runtime.h>

// ---------------------------------------------------------------------------
// GloVe loss on MI455X (gfx1250).
// Batched 300-dim f32 dot products via V_WMMA_F32_16X16X4_F32 (diagonal of a
// 16x16 accumulator = 16 pair-dots per wave). Memory-bound by L2 gathers;
// 16384 independent waves give the MLP needed to saturate L2/HBM.
// Deterministic two-stage fixed-order reduction (no float atomics).
// ---------------------------------------------------------------------------

typedef __attribute__((ext_vector_type(2))) float v2f;
typedef __attribute__((ext_vector_type(8))) float v8f;

#define GLOVE_VOCAB 10000
#define GLOVE_EMBED 300
#define GLOVE_BATCH 262144
#define GLOVE_XMAX  100.0f
#define GLOVE_ALPHA 0.75f

#define PAIRS_PER_WAVE 16
#define WAVES_PER_BLOCK 8
#define PAIRS_PER_BLOCK (PAIRS_PER_WAVE * WAVES_PER_BLOCK)   // 128
#define NUM_BLOCKS (GLOVE_BATCH / PAIRS_PER_BLOCK)           // 2048

__global__ __launch_bounds__(256) void glove_partial_kernel(
    const int*   __restrict__ cen,     // [B] center_word_lookup
    const int*   __restrict__ ctx,     // [B] context_word_lookup
    const float* __restrict__ V,       // [VOCAB, EMBED]
    const float* __restrict__ U,       // [VOCAB, EMBED]
    const float* __restrict__ vb,      // [VOCAB]
    const float* __restrict__ ub,      // [VOCAB]
    const float* __restrict__ comat,   // [VOCAB, VOCAB]
    float*       __restrict__ partial) // [NUM_BLOCKS]
{
    const int lane = threadIdx.x & 31;
    const int wave = threadIdx.x >> 5;
    const int group = blockIdx.x * WAVES_PER_BLOCK + wave;  // 16 pairs per group
    const int base  = group * PAIRS_PER_WAVE;

    // --- fragment addressing -------------------------------------------------
    // f32 A (16x4, MxK): lane l holds M = l%16; VGPR0/1 = K = 2*(l>=16) + {0,1}
    // f32 B (4x16, KxN): lane l holds N = l%16; VGPR0/1 = K = 2*(l>=16) + {0,1}
    // With B = U-chunk transposed, both fragments read row (index[l%16]) at
    // columns kbase + 2*(l>=16) + {0,1}: one 8-byte load per lane per matrix.
    const int pslot = lane & 15;
    const int i0 = cen[base + pslot];
    const int j0 = ctx[base + pslot];
    const float* rowV = V + (long)i0 * GLOVE_EMBED;
    const float* rowU = U + (long)j0 * GLOVE_EMBED;
    const int colsel = (lane >> 4) << 1;    // 0 for lanes 0-15, 2 for lanes 16-31

    v8f c = {0.f, 0.f, 0.f, 0.f, 0.f, 0.f, 0.f, 0.f};

    // 300 = 75 * 4 steps of K. EXEC is all-ones here (WMMA requirement).
#pragma unroll 5
    for (int k = 0; k < GLOVE_EMBED; k += 4) {
        const int co = k + colsel;
        v2f a = *(const v2f*)(rowV + co);   // 8B aligned: co is even
        v2f b = *(const v2f*)(rowU + co);
        // D = A(16x4) x B(4x16) + C ; full f32 precision
        c = __builtin_amdgcn_wmma_f32_16x16x4_f32(
                /*neg_a=*/false, a, /*neg_b=*/false, b,
                /*c_mod=*/(short)0, c, /*reuse_a=*/false, /*reuse_b=*/false);
    }

    // --- diagonal extraction -------------------------------------------------
    // C/D layout: VGPR r, lane l -> M = r + 8*(l>=16), N = l%16.
    // Diagonal (m,m): lanes 0-7 have pair m = lane at c[lane];
    //                 lanes 24-31 have pair m = lane-16 at c[lane-24].
    float dot = 0.f;
    const int sel = lane & 7;
#pragma unroll
    for (int r = 0; r < 8; ++r) {
        if (sel == r) dot = c[r];
    }

    const bool active = (lane < 8) || (lane >= 24);
    float term = 0.f;
    if (active) {
        const int myp = (lane < 8) ? lane : (lane - 16);  // pair 0..15
        const int ii = cen[base + myp];
        const int jj = ctx[base + myp];
        const float x = comat[(long)ii * GLOVE_VOCAB + jj];
        const float w = (x < GLOVE_XMAX)
                            ? __powf(x * (1.0f / GLOVE_XMAX), GLOVE_ALPHA)
                            : 1.0f;
        const float resid = dot + vb[ii] + ub[jj] - __logf(x);
        term = w * resid * resid;
    }

    // --- wave reduction (fixed order -> deterministic) -----------------------
#pragma unroll
    for (int off = 16; off > 0; off >>= 1)
        term += __shfl_down(term, off, 32);

    __shared__ float sw[WAVES_PER_BLOCK];
    if (lane == 0) sw[wave] = term;
    __syncthreads();

    if (threadIdx.x == 0) {
        float s = 0.f;
#pragma unroll
        for (int w = 0; w < WAVES_PER_BLOCK; ++w) s += sw[w];
        partial[blockIdx.x] = s;
    }
}

__global__ __launch_bounds__(256) void glove_reduce_kernel(
    const float* __restrict__ partial, int n, float* __restrict__ out)
{
    __shared__ float s[256];
    float acc = 0.f;
    for (int i = threadIdx.x; i < n; i += 256) acc += partial[i];  // fixed order
    s[threadIdx.x] = acc;
    __syncthreads();
#pragma unroll
    for (int off = 128; off > 0; off >>= 1) {
        if ((int)threadIdx.x < off) s[threadIdx.x] += s[threadIdx.x + off];
        __syncthreads();
    }
    if (threadIdx.x == 0) out[0] = s[0];
}

extern "C" void kernel_launch(void* const* d_in, const int* in_sizes, int n_in,
                              void* d_out, int out_size, void* d_ws, size_t ws_size,
                              hipStream_t stream)
{
    (void)in_sizes; (void)n_in; (void)out_size; (void)ws_size;

    const int*   cen   = (const int*)  d_in[0];
    const int*   ctx   = (const int*)  d_in[1];
    const float* V     = (const float*)d_in[2];
    const float* U     = (const float*)d_in[3];
    const float* vb    = (const float*)d_in[4];
    const float* ub    = (const float*)d_in[5];
    const float* comat = (const float*)d_in[6];
    float*       out   = (float*)d_out;
    float*       ws    = (float*)d_ws;   // NUM_BLOCKS floats (8 KB)

    glove_partial_kernel<<<NUM_BLOCKS, 256, 0, stream>>>(
        cen, ctx, V, U, vb, ub, comat, ws);
    glove_reduce_kernel<<<1, 256, 0, stream>>>(ws, NUM_BLOCKS, out);
}